// GATClassifier_58256936403163
// MI455X (gfx1250) — compile-verified
//
#include <hip/hip_runtime.h>

// ---------------- problem constants ----------------
#define NN   50000      // nodes (multiple of 16)
#define EE   800000     // edges
#define GG   64         // graphs
#define HH   3          // heads
#define CC   64         // channels / head
#define HC   192        // H*C
#define KIN  75         // input feature dim
#define ENL  (EE + NN)  // edges + self loops
#define HC4  (HC / 4)   // 48 quads per node row

typedef __attribute__((ext_vector_type(2))) float        v2f;
typedef __attribute__((ext_vector_type(8))) float        v8f;
typedef __attribute__((ext_vector_type(4))) unsigned int u32x4;
typedef __attribute__((ext_vector_type(4))) int          i32x4;
typedef __attribute__((ext_vector_type(8))) int          i32x8;

#if __has_include(<hip/amd_detail/amd_gfx1250_TDM.h>)
#define TDM_6ARG 1
#endif

// ---------------- helpers ----------------
__device__ __forceinline__ void atomicMaxF(float* addr, float val) {
  unsigned int* ua = (unsigned int*)addr;
  unsigned int old = *ua;
  while (true) {
    float cur = __uint_as_float(old);
    if (cur >= val) break;
    unsigned int assumed = old;
    old = atomicCAS(ua, assumed, __float_as_uint(val));
    if (old == assumed) break;
  }
}

// select-based guarded load (clamped address, cndmask result): no EXEC games
__device__ __forceinline__ float ldg_guard(const float* __restrict__ p, int idx, bool ok) {
  float v = p[ok ? idx : 0];
  return ok ? v : 0.0f;
}

// ---------------- generic fill ----------------
__global__ void fill_f32(float* p, float v, int n) {
  int i = blockIdx.x * blockDim.x + threadIdx.x;
  if (i < n) p[i] = v;
}

// ---------------- fp32 WMMA GEMM: Out[M,Nn] = A[M,K] @ B[K,Nn] ----------------
// Block = 8 waves, one N-tile per block (blockIdx.y), 8 consecutive M-tiles.
// B panel (K x 16) staged into LDS once per block via the Tensor Data Mover,
// then V_WMMA_F32_16X16X4_F32 with A fragments from global, B from LDS.
// Requires M % 16 == 0 (true here: 50000).
__global__ __launch_bounds__(256)
void gemm_f32_wmma(const float* __restrict__ A, const float* __restrict__ B,
                   float* __restrict__ Out, int M, int K, int Nn) {
  __shared__ float ldsB[HC * 16];                // up to 192x16 fp32 = 12 KB
  const int n0 = blockIdx.y << 4;

  // ---- stage B[0:K, n0:n0+16] into LDS ----
#if defined(__has_builtin) && __has_builtin(__builtin_amdgcn_tensor_load_to_lds)
  if (threadIdx.x < 32) {                        // wave 0 issues the TDM op
    unsigned long long gaddr = (unsigned long long)(const void*)(B + n0);
    u32x4 g0;
    g0[0] = 1u;                                  // count = 1 (valid descriptor)
    g0[1] = 0u;                                  // lds_addr = 0 (ldsB @ offset 0)
    g0[2] = (unsigned)(gaddr & 0xffffffffu);     // global_addr[31:0]
    g0[3] = (unsigned)((gaddr >> 32) & 0x1ffffffu) | (2u << 30);  // addr hi | type=2
    i32x8 g1;
    g1[0] = (int)(2u << 16);                     // wg_mask=0, data_size=2 (4B)
    g1[1] = (int)((unsigned)(Nn & 0xffff) << 16);          // tensor_dim0 lo @ [63:48]
    g1[2] = (int)(((unsigned)Nn >> 16) | ((unsigned)(K & 0xffff) << 16)); // dim0 hi | dim1 lo
    g1[3] = (int)((((unsigned)K >> 16) & 0xffffu) | (16u << 16));         // dim1 hi | tile_dim0=16
    g1[4] = (int)(K & 0xffff);                   // tile_dim1 = K, tile_dim2 = 0
    g1[5] = (int)Nn;                             // tensor_dim0_stride[31:0]
    g1[6] = 0;                                   // stride hi, dim1_stride lo
    g1[7] = 0;
    i32x4 z4 = {0, 0, 0, 0};
#ifdef TDM_6ARG
    i32x8 z8 = {0, 0, 0, 0, 0, 0, 0, 0};
    __builtin_amdgcn_tensor_load_to_lds(g0, g1, z4, z4, z8, 0);
#else
    __builtin_amdgcn_tensor_load_to_lds(g0, g1, z4, z4, 0);
#endif
    __builtin_amdgcn_s_wait_tensorcnt(0);
  }
#else
  for (int idx = threadIdx.x; idx < K * 16; idx += blockDim.x)
    ldsB[idx] = B[(idx >> 4) * Nn + n0 + (idx & 15)];
#endif
  __syncthreads();
  // The TDM wrote ldsB behind the compiler's back (builtin only consumes the
  // descriptor values). Escape the pointer + memory clobber so the reads
  // below are real ds_loads instead of being folded to undef.
  asm volatile("" : : "r"(ldsB) : "memory");

  const int lane = threadIdx.x & 31;
  const int mt   = blockIdx.x * 8 + (threadIdx.x >> 5);
  if (mt >= (M >> 4)) return;                    // wave-uniform exit (after barrier)
  const int m0   = mt << 4;
  const int half = lane >> 4;                    // 0: K+0/1   1: K+2/3
  const int l16  = lane & 15;
  const int am   = m0 + l16;                     // A row for this lane (always < M)
  const int K4   = K & ~3;

  v8f acc = {};
  for (int k = 0; k < K4; k += 4) {              // unguarded steady state
    const int ka = k + half * 2;
    v2f a, b;
    a.x = A[am * K + ka];
    a.y = A[am * K + ka + 1];
    b.x = ldsB[ka * 16 + l16];
    b.y = ldsB[(ka + 1) * 16 + l16];
    acc = __builtin_amdgcn_wmma_f32_16x16x4_f32(
        false, a, false, b, (short)0, acc, false, false);
  }
  if (K4 < K) {                                  // ragged K tail (layer 1: K=75)
    const int ka = K4 + half * 2;
    const bool ok0 = (ka < K), ok1 = (ka + 1 < K);
    v2f a, b;
    a.x = ldg_guard(A, am * K + ka, ok0);
    a.y = ldg_guard(A, am * K + ka + 1, ok1);
    b.x = ok0 ? ldsB[ka * 16 + l16] : 0.0f;      // LDS array covers index range
    b.y = ok1 ? ldsB[(ka + 1) * 16 + l16] : 0.0f;
    acc = __builtin_amdgcn_wmma_f32_16x16x4_f32(
        false, a, false, b, (short)0, acc, false, false);
  }
  // C/D layout: VGPR r -> row m0 + half*8 + r, col n0 + l16
  #pragma unroll
  for (int r = 0; r < 8; ++r)
    Out[(m0 + half * 8 + r) * Nn + n0 + l16] = acc[r];
}

// ---------------- degree + edge-attr segment sums (self-loop fill) ----------
__global__ void deg_sum_k(const int* __restrict__ dst, const float* __restrict__ ea,
                          float* cnt, float* sattr) {
  int e = blockIdx.x * blockDim.x + threadIdx.x;
  if (e >= EE) return;
  int d = dst[e];
  atomicAdd(&cnt[d], 1.0f);
  atomicAdd(&sattr[d], ea[e]);
}

__global__ void loop_attr_k(const float* __restrict__ cnt, const float* __restrict__ sattr,
                            float* la) {
  int n = blockIdx.x * blockDim.x + threadIdx.x;
  if (n < NN) la[n] = sattr[n] / fmaxf(cnt[n], 1.0f);
}

// ---------------- per-node attention dot products ---------------------------
__global__ void attvec_k(const float* __restrict__ xh,
                         const float* __restrict__ a_src, const float* __restrict__ a_dst,
                         float* asrc, float* adst) {
  int t = blockIdx.x * blockDim.x + threadIdx.x;     // over N*H
  if (t >= NN * HH) return;
  int n = t / HH, h = t % HH;
  const float* xv = xh + n * HC + h * CC;
  const float* sv = a_src + h * CC;
  const float* dv = a_dst + h * CC;
  float ss = 0.f, dd = 0.f;
  for (int c = 0; c < CC; c += 4) {
    float4 x = *(const float4*)(xv + c);
    float4 s = *(const float4*)(sv + c);
    float4 d = *(const float4*)(dv + c);
    ss += x.x * s.x + x.y * s.y + x.z * s.z + x.w * s.w;
    dd += x.x * d.x + x.y * d.y + x.z * d.z + x.w * d.w;
  }
  asrc[t] = ss; adst[t] = dd;
}

// we_ae[h] = dot(We[0, h*C:(h+1)*C], a_e[h])  (ED == 1)
__global__ void weae_k(const float* __restrict__ We, const float* __restrict__ ae, float* weae) {
  int h = threadIdx.x;
  if (h < HH) {
    float a = 0.f;
    for (int c = 0; c < CC; ++c) a += We[h * CC + c] * ae[h * CC + c];
    weae[h] = a;
  }
}

// ---------------- alpha = lrelu(asrc[s]+adst[d]+ea*we_ae) + segment max -----
__global__ void alpha_k(const int* __restrict__ src, const int* __restrict__ dst,
                        const float* __restrict__ ea, const float* __restrict__ la,
                        const float* __restrict__ asrc, const float* __restrict__ adst,
                        const float* __restrict__ weae,
                        float* albuf, float* amax) {
  int i = blockIdx.x * blockDim.x + threadIdx.x;
  if (i >= ENL) return;
  int s, d; float e;
  if (i < EE) { s = src[i]; d = dst[i]; e = ea[i]; }
  else        { s = d = i - EE;         e = la[s]; }
  #pragma unroll
  for (int h = 0; h < HH; ++h) {
    float a = asrc[s * HH + h] + adst[d * HH + h] + e * weae[h];
    a = a > 0.f ? a : 0.2f * a;                    // leaky_relu(0.2)
    albuf[i * HH + h] = a;
    atomicMaxF(&amax[d * HH + h], a);
  }
}

__global__ void exp_k(const int* __restrict__ dst, float* albuf,
                      const float* __restrict__ amax, float* denom) {
  int i = blockIdx.x * blockDim.x + threadIdx.x;
  if (i >= ENL) return;
  int d = (i < EE) ? dst[i] : (i - EE);
  #pragma unroll
  for (int h = 0; h < HH; ++h) {
    float ex = __expf(albuf[i * HH + h] - amax[d * HH + h]);
    albuf[i * HH + h] = ex;
    atomicAdd(&denom[d * HH + h], ex);
  }
}

__global__ void attn_k(const int* __restrict__ dst, const float* __restrict__ albuf,
                       const float* __restrict__ denom, float* attn_out) {
  int i = blockIdx.x * blockDim.x + threadIdx.x;
  if (i >= ENL) return;
  int d = (i < EE) ? dst[i] : (i - EE);
  #pragma unroll
  for (int h = 0; h < HH; ++h)
    attn_out[i * HH + h] = albuf[i * HH + h] / (denom[d * HH + h] + 1e-16f);
}

// ---------------- scatter aggregation: agg[d] += attn * xh[s] ----------------
// one thread per (edge, 4-channel quad): single b128 load + 4 atomics
__global__ void agg_k(const int* __restrict__ src, const int* __restrict__ dst,
                      const float* __restrict__ attn, const float* __restrict__ xh,
                      float* agg) {
  long t = (long)blockIdx.x * blockDim.x + threadIdx.x;
  if (t >= (long)ENL * HC4) return;
  int i  = (int)(t / HC4);
  int hc = (int)(t % HC4) * 4;
  int h  = hc >> 6;
  int s, d;
  if (i < EE) { s = src[i]; d = dst[i]; } else { s = d = i - EE; }
  float  a  = attn[i * HH + h];
  float4 xv = *(const float4*)(xh + s * HC + hc);
  float* o  = agg + d * HC + hc;
  atomicAdd(o + 0, a * xv.x);
  atomicAdd(o + 1, a * xv.y);
  atomicAdd(o + 2, a * xv.z);
  atomicAdd(o + 3, a * xv.w);
}

__global__ void bias_elu_k(float* agg, const float* __restrict__ b) {
  int t = blockIdx.x * blockDim.x + threadIdx.x;   // over N*HC4
  if (t >= NN * HC4) return;
  int base = t * 4, hc = base % HC;
  float4 v  = *(float4*)(agg + base);
  float4 bb = *(const float4*)(b + hc);
  v.x += bb.x; v.y += bb.y; v.z += bb.z; v.w += bb.w;
  v.x = v.x > 0.f ? v.x : (__expf(v.x) - 1.f);
  v.y = v.y > 0.f ? v.y : (__expf(v.y) - 1.f);
  v.z = v.z > 0.f ? v.z : (__expf(v.z) - 1.f);
  v.w = v.w > 0.f ? v.w : (__expf(v.w) - 1.f);
  *(float4*)(agg + base) = v;
}

// ---------------- global mean/max pool per graph -----------------------------
__global__ void pool_k(const float* __restrict__ h, const int* __restrict__ batch,
                       float* gsum, float* gmax, float* gcnt) {
  int t = blockIdx.x * blockDim.x + threadIdx.x;   // over N*HC4
  if (t >= NN * HC4) return;
  int n = t / HC4, q = t % HC4, hc = q * 4;
  int g = batch[n];
  float4 v = *(const float4*)(h + n * HC + hc);
  float* sp = gsum + g * HC + hc;
  float* mp = gmax + g * HC + hc;
  atomicAdd(sp + 0, v.x); atomicMaxF(mp + 0, v.x);
  atomicAdd(sp + 1, v.y); atomicMaxF(mp + 1, v.y);
  atomicAdd(sp + 2, v.z); atomicMaxF(mp + 2, v.z);
  atomicAdd(sp + 3, v.w); atomicMaxF(mp + 3, v.w);
  if (q == 0) atomicAdd(&gcnt[g], 1.0f);
}

// ---------------- MLP head: relu(z@fc1+b1) @ fc2 + b2 ------------------------
__global__ void mlp_k(const float* __restrict__ gsum, const float* __restrict__ gmax,
                      const float* __restrict__ gcnt,
                      const float* __restrict__ w1, const float* __restrict__ b1,
                      const float* __restrict__ w2, const float* __restrict__ b2,
                      float* logits) {
  int g = threadIdx.x;
  if (g >= GG) return;
  float inv = 1.0f / fmaxf(gcnt[g], 1.0f);
  float acc = 0.f;
  for (int j = 0; j < CC; ++j) {
    float hsum = b1[j];
    for (int k = 0; k < HC; ++k) hsum += (gsum[g * HC + k] * inv) * w1[k * CC + j];
    for (int k = 0; k < HC; ++k) hsum += gmax[g * HC + k] * w1[(HC + k) * CC + j];
    hsum = fmaxf(hsum, 0.f);
    acc += hsum * w2[j];
  }
  logits[g] = acc + b2[0];
}

// =============================================================================
extern "C" void kernel_launch(void* const* d_in, const int* in_sizes, int n_in,
                              void* d_out, int out_size, void* d_ws, size_t ws_size,
                              hipStream_t stream) {
  (void)in_sizes; (void)n_in; (void)out_size; (void)ws_size;

  const float* x    = (const float*)d_in[0];
  const float* eatt = (const float*)d_in[1];
  const float* W1   = (const float*)d_in[2];
  const float* as1  = (const float*)d_in[3];
  const float* ad1  = (const float*)d_in[4];
  const float* We1  = (const float*)d_in[5];
  const float* ae1  = (const float*)d_in[6];
  const float* b1   = (const float*)d_in[7];
  const float* W2   = (const float*)d_in[8];
  const float* as2  = (const float*)d_in[9];
  const float* ad2  = (const float*)d_in[10];
  const float* We2  = (const float*)d_in[11];
  const float* ae2  = (const float*)d_in[12];
  const float* b2   = (const float*)d_in[13];
  const float* fc1w = (const float*)d_in[14];
  const float* fc1b = (const float*)d_in[15];
  const float* fc2w = (const float*)d_in[16];
  const float* fc2b = (const float*)d_in[17];
  const int*   eidx = (const int*)d_in[18];
  const int*   batch= (const int*)d_in[19];
  const int* src = eidx;
  const int* dst = eidx + EE;

  // outputs: (logits[G], attn1[(E+N)*H], attn2[(E+N)*H]) flattened in order
  float* out    = (float*)d_out;
  float* logits = out;
  float* attn1  = out + GG;
  float* attn2  = attn1 + (long)ENL * HH;

  // workspace layout
  float* ws    = (float*)d_ws;
  float* xh    = ws;              ws += NN * HC;     // per-layer x@W
  float* agg   = ws;              ws += NN * HC;     // aggregation / hidden
  float* albuf = ws;              ws += ENL * HH;    // alpha / exp scratch
  float* asrcB = ws;              ws += NN * HH;
  float* adstB = ws;              ws += NN * HH;
  float* amax  = ws;              ws += NN * HH;
  float* denom = ws;              ws += NN * HH;
  float* cnt   = ws;              ws += NN;
  float* sattr = ws;              ws += NN;
  float* la    = ws;              ws += NN;
  float* weae  = ws;              ws += 4;
  float* gsum  = ws;              ws += GG * HC;
  float* gmax  = ws;              ws += GG * HC;
  float* gcnt  = ws;              ws += GG;

  const int TB = 256;
  auto blk = [](long n) { return (int)((n + 255) / 256); };
  const dim3 gemm_grid((NN / 16 + 7) / 8, HC / 16);  // 391 x 12 blocks, 8 waves each
  const float NEGINF = -3.4e38f;

  // ---- self-loop edge attr (shared by both layers) ----
  fill_f32<<<blk(NN), TB, 0, stream>>>(cnt, 0.f, NN);
  fill_f32<<<blk(NN), TB, 0, stream>>>(sattr, 0.f, NN);
  deg_sum_k<<<blk(EE), TB, 0, stream>>>(dst, eatt, cnt, sattr);
  loop_attr_k<<<blk(NN), TB, 0, stream>>>(cnt, sattr, la);

  // ================= layer 1 =================
  gemm_f32_wmma<<<gemm_grid, TB, 0, stream>>>(x, W1, xh, NN, KIN, HC);
  weae_k<<<1, 32, 0, stream>>>(We1, ae1, weae);
  attvec_k<<<blk(NN * HH), TB, 0, stream>>>(xh, as1, ad1, asrcB, adstB);
  fill_f32<<<blk(NN * HH), TB, 0, stream>>>(amax, NEGINF, NN * HH);
  fill_f32<<<blk(NN * HH), TB, 0, stream>>>(denom, 0.f, NN * HH);
  alpha_k<<<blk(ENL), TB, 0, stream>>>(src, dst, eatt, la, asrcB, adstB, weae, albuf, amax);
  exp_k<<<blk(ENL), TB, 0, stream>>>(dst, albuf, amax, denom);
  attn_k<<<blk(ENL), TB, 0, stream>>>(dst, albuf, denom, attn1);
  fill_f32<<<blk((long)NN * HC), TB, 0, stream>>>(agg, 0.f, NN * HC);
  agg_k<<<blk((long)ENL * HC4), TB, 0, stream>>>(src, dst, attn1, xh, agg);
  bias_elu_k<<<blk((long)NN * HC4), TB, 0, stream>>>(agg, b1);      // agg = h1

  // ================= layer 2 =================
  gemm_f32_wmma<<<gemm_grid, TB, 0, stream>>>(agg, W2, xh, NN, HC, HC);
  weae_k<<<1, 32, 0, stream>>>(We2, ae2, weae);
  attvec_k<<<blk(NN * HH), TB, 0, stream>>>(xh, as2, ad2, asrcB, adstB);
  fill_f32<<<blk(NN * HH), TB, 0, stream>>>(amax, NEGINF, NN * HH);
  fill_f32<<<blk(NN * HH), TB, 0, stream>>>(denom, 0.f, NN * HH);
  alpha_k<<<blk(ENL), TB, 0, stream>>>(src, dst, eatt, la, asrcB, adstB, weae, albuf, amax);
  exp_k<<<blk(ENL), TB, 0, stream>>>(dst, albuf, amax, denom);
  attn_k<<<blk(ENL), TB, 0, stream>>>(dst, albuf, denom, attn2);
  fill_f32<<<blk((long)NN * HC), TB, 0, stream>>>(agg, 0.f, NN * HC); // h1 consumed
  agg_k<<<blk((long)ENL * HC4), TB, 0, stream>>>(src, dst, attn2, xh, agg);
  bias_elu_k<<<blk((long)NN * HC4), TB, 0, stream>>>(agg, b2);      // agg = h2

  // ================= pooling + MLP head =================
  fill_f32<<<blk(GG * HC), TB, 0, stream>>>(gsum, 0.f, GG * HC);
  fill_f32<<<blk(GG * HC), TB, 0, stream>>>(gmax, NEGINF, GG * HC);
  fill_f32<<<blk(GG), TB, 0, stream>>>(gcnt, 0.f, GG);
  pool_k<<<blk((long)NN * HC4), TB, 0, stream>>>(agg, batch, gsum, gmax, gcnt);
  mlp_k<<<1, 64, 0, stream>>>(gsum, gmax, gcnt, fc1w, fc1b, fc2w, fc2b, logits);
}